// YOLOLoss_88184268521539
// MI455X (gfx1250) — compile-verified
//
#include <hip/hip_runtime.h>
#include <hip/hip_bf16.h>
#include <math.h>

typedef float v2f __attribute__((ext_vector_type(2)));
typedef float v8f __attribute__((ext_vector_type(8)));

#define EPSF 1e-6f
__device__ __forceinline__ float sigmoidf_(float x) { return 1.0f / (1.0f + expf(-x)); }

__device__ __forceinline__ float bcef_(float p, float t) {
    p = fminf(fmaxf(p, 1e-7f), 1.0f - 1e-7f);
    return -t * logf(p) - (1.0f - t) * logf(1.0f - p);
}

// Exact fp32 sum of one value per lane across the wave32, via V_WMMA_F32_16X16X4_F32.
// A holds the 32 lane values (a.y=0 zeroes the other 32 slots), B = ones(4x16).
// D[m][n] = rowsum_A(m); one COLUMN of D sums to the wave total. Lane L holds
// rows 0..7 of column L, lane L^16 rows 8..15 -> add d[0..7] then xor-16 shuffle.
__device__ __forceinline__ float wave_sum32(float v) {
    v2f a; a.x = v; a.y = 0.0f;
    v2f b; b.x = 1.0f; b.y = 1.0f;
    v8f c = {0.f,0.f,0.f,0.f,0.f,0.f,0.f,0.f};
    v8f d = __builtin_amdgcn_wmma_f32_16x16x4_f32(false, a, false, b, (short)0, c, false, false);
    float s = d[0] + d[1] + d[2] + d[3] + d[4] + d[5] + d[6] + d[7];
    s += __shfl_xor(s, 16, 32);
    return s;
}

// Sutherland-Hodgman clip of rect1 against rect2; returns skew IoU (reference semantics).
__device__ float skew_iou_dev(float x1, float y1, float w1, float h1, float a1,
                              float x2, float y2, float w2, float h2, float a2) {
    float c1x[4], c1y[4], c2x[4], c2y[4];
    {
        float ca = cosf(a1), sa = sinf(a1);
        float dx[4] = {0.5f*w1, -0.5f*w1, -0.5f*w1, 0.5f*w1};
        float dy[4] = {0.5f*h1, 0.5f*h1, -0.5f*h1, -0.5f*h1};
        #pragma unroll
        for (int k = 0; k < 4; ++k) { c1x[k] = x1 + dx[k]*ca - dy[k]*sa; c1y[k] = y1 + dx[k]*sa + dy[k]*ca; }
    }
    {
        float ca = cosf(a2), sa = sinf(a2);
        float dx[4] = {0.5f*w2, -0.5f*w2, -0.5f*w2, 0.5f*w2};
        float dy[4] = {0.5f*h2, 0.5f*h2, -0.5f*h2, -0.5f*h2};
        #pragma unroll
        for (int k = 0; k < 4; ++k) { c2x[k] = x2 + dx[k]*ca - dy[k]*sa; c2y[k] = y2 + dx[k]*sa + dy[k]*ca; }
    }
    float plx[8], ply[8];
    int n = 4;
    #pragma unroll
    for (int k = 0; k < 4; ++k) { plx[k] = c1x[k]; ply[k] = c1y[k]; }

    for (int j = 0; j < 4; ++j) {
        float ax = c2x[j], ay = c2y[j];
        float bx = c2x[(j + 1) & 3], by = c2y[(j + 1) & 3];
        float ex = bx - ax, ey = by - ay;
        float nx[16], ny[16];
        int m = 0;
        for (int k = 0; k < n; ++k) {
            float pxk = plx[k], pyk = ply[k];
            int k1 = (k + 1 == n) ? 0 : k + 1;
            float qx = plx[k1], qy = ply[k1];
            float dp = ex * (pyk - ay) - ey * (pxk - ax);
            float dq = ex * (qy - ay) - ey * (qx - ax);
            bool inp = dp >= 0.0f, inq = dq >= 0.0f;
            if (inp) { nx[m] = pxk; ny[m] = pyk; ++m; }
            if (inp != inq) {
                float denom = dp - dq;
                float t = dp / ((fabsf(denom) < 1e-12f) ? 1.0f : denom);
                nx[m] = pxk + t * (qx - pxk);
                ny[m] = pyk + t * (qy - pyk);
                ++m;
            }
        }
        n = (m > 8) ? 8 : m;               // reference keeps first 8 candidates
        for (int k = 0; k < n; ++k) { plx[k] = nx[k]; ply[k] = ny[k]; }
        if (n == 0) break;
    }
    float area = 0.0f;
    for (int k = 0; k < n; ++k) {
        int k1 = (k + 1 == n) ? 0 : k + 1;
        area += plx[k] * ply[k1] - plx[k1] * ply[k];
    }
    area = 0.5f * fabsf(area);
    float ar1 = fabsf(w1 * h1), ar2 = fabsf(w2 * h2);
    return area / (ar1 + ar2 - area + 1e-9f);
}

__global__ __launch_bounds__(256)
void yolo_loss_main(const float* __restrict__ inp, const float* __restrict__ yt,
                    const float* __restrict__ noobj, const float* __restrict__ blsc,
                    const float* __restrict__ anchors,
                    int cells, int A, int attrs, int H, int W,
                    float* __restrict__ partials) {
    const int i = blockIdx.x * blockDim.x + threadIdx.x;
    float s_mf = 0.f, s_loc = 0.f, s_cpos = 0.f, s_cnoo = 0.f, s_cls = 0.f;

    if (i < cells) {
        const long HW = (long)H * W;
        const int w = i % W;
        const int rem = i / W;
        const int h = rem % H;
        const int ba = rem / H;           // b*A + a
        const int a = ba % A;
        const float* ycell = yt + (size_t)i * attrs;
        const float tobj = ycell[5];
        const long off = (long)ba * attrs * HW + (long)h * W + w;

        const float conf = sigmoidf_(inp[off + 5 * HW]);
        const float bce_c = bcef_(conf, tobj);
        s_cnoo = bce_c * noobj[i];
        s_mf = tobj;

        if (tobj == 1.0f) {               // positive cell (~0.3%): full loss path
            s_cpos = bce_c;
            const float aw = anchors[a * 3 + 0];
            const float ah = anchors[a * 3 + 1];
            const float aa = anchors[a * 3 + 2];
            const float px = sigmoidf_(inp[off]) + (float)w;
            const float py = sigmoidf_(inp[off + HW]) + (float)h;
            const float pw = expf(inp[off + 2 * HW]) * aw;
            const float ph = expf(inp[off + 3 * HW]) * ah;
            const float pang = inp[off + 4 * HW] + aa;
            const float tx = ycell[0], ty = ycell[1], tw = ycell[2], th = ycell[3], tang = ycell[4];

            float cls = 0.f;
            for (int k = 6; k < attrs; ++k)
                cls += bcef_(sigmoidf_(inp[off + (long)k * HW]), ycell[k]);
            s_cls = cls;

            // CIoU
            const float b1nx = px - 0.5f * pw, b1xx = px + 0.5f * pw;
            const float b1ny = py - 0.5f * ph, b1xy = py + 0.5f * ph;
            const float b2nx = tx - 0.5f * tw, b2xx = tx + 0.5f * tw;
            const float b2ny = ty - 0.5f * th, b2xy = ty + 0.5f * th;
            const float iwx = fmaxf(fminf(b1xx, b2xx) - fmaxf(b1nx, b2nx), 0.f);
            const float iwy = fmaxf(fminf(b1xy, b2xy) - fmaxf(b1ny, b2ny), 0.f);
            const float inter = iwx * iwy;
            const float uni = pw * ph + tw * th - inter;
            const float iou = inter / fmaxf(uni, EPSF);
            const float cd = (px - tx) * (px - tx) + (py - ty) * (py - ty);
            const float ewx = fmaxf(fmaxf(b1xx, b2xx) - fminf(b1nx, b2nx), 0.f);
            const float ewy = fmaxf(fmaxf(b1xy, b2xy) - fminf(b1ny, b2ny), 0.f);
            const float diag = ewx * ewx + ewy * ewy;
            float ciou = iou - cd / fmaxf(diag, EPSF);
            const float dv = atanf(pw / fmaxf(ph, EPSF)) - atanf(tw / fmaxf(th, EPSF));
            const float v = 0.40528473456935108577f * dv * dv;   // 4/pi^2
            const float alpha = v / fmaxf(1.0f - iou + v, EPSF);
            ciou -= alpha * v;
            const float ciou_loss = (1.0f - ciou) * (2.0f - blsc[i]);

            const float dang = pang - tang;
            const float adang = fabsf(dang);
            const float aloss = (adang < 1.0f) ? 0.5f * dang * dang : adang - 0.5f;
            const float reg = aloss + ciou_loss;

            const float skew = skew_iou_dev(px, py, pw, ph, pang, tx, ty, tw, th, tang);
            const float skew_loss = expf(1.0f - skew) - 1.0f;
            const float rc = skew_loss / reg;    // reference: div then mul (stop_gradient)
            s_loc = reg * rc;
        }
    }

    // block reduction: wave32 WMMA sums, then cross-wave via LDS (deterministic)
    __shared__ float wsum[8][5];
    const int wave = threadIdx.x >> 5;
    const int lane = threadIdx.x & 31;
    const float r0 = wave_sum32(s_mf);
    const float r1 = wave_sum32(s_loc);
    const float r2 = wave_sum32(s_cpos);
    const float r3 = wave_sum32(s_cnoo);
    const float r4 = wave_sum32(s_cls);
    if (lane == 0) {
        wsum[wave][0] = r0; wsum[wave][1] = r1; wsum[wave][2] = r2;
        wsum[wave][3] = r3; wsum[wave][4] = r4;
    }
    __syncthreads();
    if (threadIdx.x == 0) {
        float t0 = 0.f, t1 = 0.f, t2 = 0.f, t3 = 0.f, t4 = 0.f;
        #pragma unroll
        for (int q = 0; q < 8; ++q) {
            t0 += wsum[q][0]; t1 += wsum[q][1]; t2 += wsum[q][2];
            t3 += wsum[q][3]; t4 += wsum[q][4];
        }
        float* p = partials + (size_t)blockIdx.x * 5;
        p[0] = t0; p[1] = t1; p[2] = t2; p[3] = t3; p[4] = t4;
    }
}

__global__ __launch_bounds__(256)
void yolo_loss_final(const float* __restrict__ partials, int nblocks, float* __restrict__ out) {
    __shared__ float red[5 * 256];
    float s[5] = {0.f, 0.f, 0.f, 0.f, 0.f};
    for (int b = threadIdx.x; b < nblocks; b += 256) {
        #pragma unroll
        for (int q = 0; q < 5; ++q) s[q] += partials[(size_t)b * 5 + q];
    }
    #pragma unroll
    for (int q = 0; q < 5; ++q) red[q * 256 + threadIdx.x] = s[q];
    __syncthreads();
    for (int stride = 128; stride > 0; stride >>= 1) {
        if ((int)threadIdx.x < stride) {
            #pragma unroll
            for (int q = 0; q < 5; ++q)
                red[q * 256 + threadIdx.x] += red[q * 256 + threadIdx.x + stride];
        }
        __syncthreads();
    }
    if (threadIdx.x == 0) {
        const float smf = red[0], sloc = red[256], scpos = red[512], scnoo = red[768], scls = red[1024];
        const float npos = fmaxf(smf, 1.0f);
        const float lloc = sloc / npos;
        const float lconf = scpos / npos + scnoo;
        const float lcls = scls;
        out[0] = lloc + lconf + lcls;
        out[1] = smf;
        out[2] = lloc;
        out[3] = lconf;
        out[4] = lcls;
    }
}

extern "C" void kernel_launch(void* const* d_in, const int* in_sizes, int n_in,
                              void* d_out, int out_size, void* d_ws, size_t ws_size,
                              hipStream_t stream) {
    const float* inp     = (const float*)d_in[0];  // (bs, A*attrs, H, W)
    const float* ytrue   = (const float*)d_in[1];  // (bs, A, H, W, attrs)
    const float* noobj   = (const float*)d_in[2];  // (bs, A, H, W)
    const float* blsc    = (const float*)d_in[3];  // (bs, A, H, W)
    const float* anchors = (const float*)d_in[4];  // (A, 3)

    const int cells = in_sizes[2];           // bs*A*H*W
    const int A     = in_sizes[4] / 3;
    const int attrs = in_sizes[0] / cells;   // C/A = 6 + ncls
    const int H = 76, W = 76;                // reference grid

    const int threads = 256;
    const int blocks = (cells + threads - 1) / threads;
    float* partials = (float*)d_ws;          // blocks * 5 floats

    yolo_loss_main<<<blocks, threads, 0, stream>>>(inp, ytrue, noobj, blsc, anchors,
                                                   cells, A, attrs, H, W, partials);
    yolo_loss_final<<<1, 256, 0, stream>>>(partials, blocks, (float*)d_out);
}